// GNN_node_74955769250250
// MI455X (gfx1250) — compile-verified
//
#include <hip/hip_runtime.h>
#include <hip/hip_bf16.h>

typedef __attribute__((ext_vector_type(2))) float v2f;
typedef __attribute__((ext_vector_type(8))) float v8f;

#define F 128
#define ROWS 64          // rows per workgroup in the fused MLP kernel
#define LDS_STRIDE 132   // pad rows: bank = (4*r + k) % 64 -> conflict-free A reads
#define BN_EPS 1e-5f

// ---------------------------------------------------------------------------
// Fold eval-mode BN (running stats = 0/1) into weights + bias:
//   BN(z@W + b) = z @ (W * s) + (b * s + beta),  s = gamma / sqrt(1 + eps)
// W is [128 x 128] row-major (k, n). 16384 threads.
// ---------------------------------------------------------------------------
__global__ __launch_bounds__(256) void fold_bn_kernel(
    const float* __restrict__ W, const float* __restrict__ b,
    const float* __restrict__ g, const float* __restrict__ be,
    float* __restrict__ Wf, float* __restrict__ bf) {
  int i = blockIdx.x * blockDim.x + threadIdx.x;   // 0..16383
  int n = i & (F - 1);
  float s = g[n] * __frsqrt_rn(1.0f + BN_EPS);
  Wf[i] = W[i] * s;
  if (i < F) bf[i] = b[i] * s + be[i];
}

// z = h  (the (1+eps)*h_i term of GIN with eps=0), float4-vectorized
__global__ __launch_bounds__(256) void copy_kernel(
    const float* __restrict__ h, float* __restrict__ z, int n4) {
  int i = blockIdx.x * blockDim.x + threadIdx.x;
  if (i < n4) ((float4*)z)[i] = ((const float4*)h)[i];
}

// z[dst] += h[src] over all edges. One wave per edge; each lane owns a
// 4-float feature chunk (32 lanes x 4 = 128). src/dst are wave-uniform so
// the compiler scalarizes the index loads.
__global__ __launch_bounds__(256) void scatter_add_kernel(
    const float* __restrict__ h, const int* __restrict__ src,
    const int* __restrict__ dst, float* __restrict__ z, int total) {
  int idx = blockIdx.x * blockDim.x + threadIdx.x;
  if (idx >= total) return;
  int e = idx >> 5;
  int c = (idx & 31) << 2;
  int s = src[e];
  int d = dst[e];
  float4 v = *(const float4*)(h + (size_t)s * F + c);
  float* zp = z + (size_t)d * F + c;
  atomicAdd(zp + 0, v.x);
  atomicAdd(zp + 1, v.y);
  atomicAdd(zp + 2, v.z);
  atomicAdd(zp + 3, v.w);
}

// ---------------------------------------------------------------------------
// Fused GIN MLP: out = [relu]( bn2( relu(bn1(z@W1+b1)) @ W2 + b2 ) )
// BN already folded into Wf1/bf1, Wf2/bf2.
// Block: 256 threads (8 waves), 64 rows. Wave w owns column tile n0 = 16*w
// and all four 16-row tiles (4 independent accumulator chains). The 64x128
// intermediate t never leaves LDS. V_WMMA_F32_16X16X4_F32 K-loop.
// ---------------------------------------------------------------------------
__global__ __launch_bounds__(256) void gin_mlp_kernel(
    const float* __restrict__ z,
    const float* __restrict__ Wf1, const float* __restrict__ bf1,
    const float* __restrict__ Wf2, const float* __restrict__ bf2,
    float* __restrict__ out, int do_relu, int Nrows) {
  __shared__ float zs[ROWS][LDS_STRIDE];
  __shared__ float ts[ROWS][LDS_STRIDE];

  const int row0 = blockIdx.x * ROWS;
  const int tid  = threadIdx.x;

  // ---- stage 64x128 z block into LDS (256 thr x 8 float4) ----
#pragma unroll
  for (int i = 0; i < ROWS * (F / 4) / 256; ++i) {
    int idx = tid + i * 256;            // float4 slots
    int r = idx >> 5;                   // 0..63
    int c = (idx & 31) << 2;            // 0..124
    int rg = row0 + r;
    if (rg >= Nrows) rg = Nrows - 1;    // clamp for the ragged last block
    float4 v = *(const float4*)(z + (size_t)rg * F + c);
    zs[r][c + 0] = v.x; zs[r][c + 1] = v.y;
    zs[r][c + 2] = v.z; zs[r][c + 3] = v.w;
  }
  __syncthreads();

  const int wave = tid >> 5;
  const int lane = tid & 31;
  const int lr   = lane & 15;           // row within A-frag / col within B-frag
  const int koff = (lane >> 4) * 2;     // upper half-wave carries K+2,K+3
  const int n0   = wave * 16;           // this wave's output-column tile
  const int rbase = (lane >> 4) * 8;    // C-matrix: VGPR v -> row v + 8*(lane>=16)

  // ===================== GEMM 1: t = relu(z @ Wf1 + bf1) =====================
  {
    float bv = bf1[n0 + lr];
    v8f acc[4];
#pragma unroll
    for (int rt = 0; rt < 4; ++rt) acc[rt] = (v8f){bv, bv, bv, bv, bv, bv, bv, bv};
    const float* wp = Wf1 + n0 + lr;
#pragma unroll 4
    for (int k = 0; k < F; k += 4) {
      v2f bfr;
      bfr.x = wp[(k + koff) * F];
      bfr.y = wp[(k + koff + 1) * F];
#pragma unroll
      for (int rt = 0; rt < 4; ++rt) {
        v2f a;
        a.x = zs[rt * 16 + lr][k + koff];
        a.y = zs[rt * 16 + lr][k + koff + 1];
        acc[rt] = __builtin_amdgcn_wmma_f32_16x16x4_f32(false, a, false, bfr,
                                                        (short)0, acc[rt], false, false);
      }
    }
#pragma unroll
    for (int rt = 0; rt < 4; ++rt) {
#pragma unroll
      for (int v = 0; v < 8; ++v) {
        float x = acc[rt][v];
        x = x > 0.0f ? x : 0.0f;
        ts[rt * 16 + rbase + v][n0 + lr] = x;
      }
    }
  }
  __syncthreads();

  // ===================== GEMM 2: out = [relu](t @ Wf2 + bf2) =================
  {
    float bv = bf2[n0 + lr];
    v8f acc[4];
#pragma unroll
    for (int rt = 0; rt < 4; ++rt) acc[rt] = (v8f){bv, bv, bv, bv, bv, bv, bv, bv};
    const float* wp = Wf2 + n0 + lr;
#pragma unroll 4
    for (int k = 0; k < F; k += 4) {
      v2f bfr;
      bfr.x = wp[(k + koff) * F];
      bfr.y = wp[(k + koff + 1) * F];
#pragma unroll
      for (int rt = 0; rt < 4; ++rt) {
        v2f a;
        a.x = ts[rt * 16 + lr][k + koff];
        a.y = ts[rt * 16 + lr][k + koff + 1];
        acc[rt] = __builtin_amdgcn_wmma_f32_16x16x4_f32(false, a, false, bfr,
                                                        (short)0, acc[rt], false, false);
      }
    }
#pragma unroll
    for (int rt = 0; rt < 4; ++rt) {
#pragma unroll
      for (int v = 0; v < 8; ++v) {
        float x = acc[rt][v];
        if (do_relu) x = x > 0.0f ? x : 0.0f;
        int r = row0 + rt * 16 + rbase + v;
        if (r < Nrows) out[(size_t)r * F + n0 + lr] = x;
      }
    }
  }
}

// ---------------------------------------------------------------------------
extern "C" void kernel_launch(void* const* d_in, const int* in_sizes, int n_in,
                              void* d_out, int out_size, void* d_ws, size_t ws_size,
                              hipStream_t stream) {
  const float* x  = (const float*)d_in[0];
  const int*   ei = (const int*)d_in[1];
  const int N = in_sizes[0] / F;
  const int E = in_sizes[1] / 2;
  const int* src = ei;       // edge_index[0]
  const int* dst = ei + E;   // edge_index[1]

  // workspace layout
  float* zbuf = (float*)d_ws;                     // N*128
  float* hbuf = zbuf + (size_t)N * F;             // N*128
  float* Wf1  = hbuf + (size_t)N * F;             // 128*128
  float* Wf2  = Wf1 + F * F;                      // 128*128
  float* bf1  = Wf2 + F * F;                      // 128
  float* bf2  = bf1 + F;                          // 128

  const int NUM_LAYERS = 4;
  const float* h = x;
  for (int layer = 0; layer < NUM_LAYERS; ++layer) {
    const float* W1  = (const float*)d_in[2 + layer * 8 + 0];
    const float* b1  = (const float*)d_in[2 + layer * 8 + 1];
    const float* g1  = (const float*)d_in[2 + layer * 8 + 2];
    const float* be1 = (const float*)d_in[2 + layer * 8 + 3];
    const float* W2  = (const float*)d_in[2 + layer * 8 + 4];
    const float* b2  = (const float*)d_in[2 + layer * 8 + 5];
    const float* g2  = (const float*)d_in[2 + layer * 8 + 6];
    const float* be2 = (const float*)d_in[2 + layer * 8 + 7];

    fold_bn_kernel<<<(F * F) / 256, 256, 0, stream>>>(W1, b1, g1, be1, Wf1, bf1);
    fold_bn_kernel<<<(F * F) / 256, 256, 0, stream>>>(W2, b2, g2, be2, Wf2, bf2);

    int n4 = N * (F / 4);
    copy_kernel<<<(n4 + 255) / 256, 256, 0, stream>>>(h, zbuf, n4);

    int total = E * 32;
    scatter_add_kernel<<<(total + 255) / 256, 256, 0, stream>>>(h, src, dst, zbuf, total);

    float* outp = (layer == NUM_LAYERS - 1) ? (float*)d_out : hbuf;
    gin_mlp_kernel<<<(N + ROWS - 1) / ROWS, 256, 0, stream>>>(
        zbuf, Wf1, bf1, Wf2, bf2, outp, (layer < NUM_LAYERS - 1) ? 1 : 0, N);
    h = hbuf;
  }
}